// TFPOLoss_87024627352244
// MI455X (gfx1250) — compile-verified
//
#include <hip/hip_runtime.h>
#include <hip/hip_bf16.h>
#include <math.h>

#define BB 8
#define SS 2048
#define NTILES (SS / 16)          // 128 m-tiles per batch row
#define TOTTILES (BB * NTILES)    // 1024 (b, m-tile) wave tasks

typedef __attribute__((ext_vector_type(2))) float v2f;
typedef __attribute__((ext_vector_type(8))) float v8f;

__device__ __forceinline__ float wave_reduce_f(float x) {
  #pragma unroll
  for (int o = 16; o > 0; o >>= 1) x += __shfl_down(x, o, 32);
  return x;
}

// ---------------------------------------------------------------------------
// Kernel 1: per batch row (one wave32 each):
//   flows = max(ps*v, eps); row sums of flow, flow^2
//   G[k]  = clip(log(flow),-10,10) + exclusive_cumsum(lp)[k]
//   H[k]  = number of mask-zeros before k  (all_mask(m,n) <=> H[m]==H[n])
// ---------------------------------------------------------------------------
__global__ void k_prep(const float* __restrict__ lp, const float* __restrict__ vals,
                       const float* __restrict__ ps, const int* __restrict__ mask,
                       float* __restrict__ G, int* __restrict__ H,
                       float* __restrict__ rowF, float* __restrict__ rowF2) {
  const int b = blockIdx.x;
  const int lane = threadIdx.x;           // 32 threads = 1 wave
  const long base = (long)b * SS;

  float carry = 0.0f;
  int zcarry = 0;
  float sF = 0.0f, sF2 = 0.0f;

  for (int c = 0; c < SS / 32; ++c) {
    const int k = c * 32 + lane;
    const float l = lp[base + k];
    const float f = fmaxf(ps[base + k] * vals[base + k], 1e-8f);
    sF += f;
    sF2 += f * f;
    float lf = logf(f);
    lf = fminf(fmaxf(lf, -10.0f), 10.0f);

    // inclusive wave scan of policy logprobs
    float incl = l;
    #pragma unroll
    for (int o = 1; o < 32; o <<= 1) {
      float t = __shfl_up(incl, o, 32);
      if (lane >= o) incl += t;
    }
    const float excl = carry + incl - l;     // exclusive prefix Cp[k]
    G[base + k] = lf + excl;
    carry += __shfl(incl, 31, 32);

    // inclusive wave scan of mask-zero indicators
    int z = (mask[base + k] == 0) ? 1 : 0;
    int iz = z;
    #pragma unroll
    for (int o = 1; o < 32; o <<= 1) {
      int t = __shfl_up(iz, o, 32);
      if (lane >= o) iz += t;
    }
    H[base + k] = zcarry + iz - z;           // zeros strictly before k
    zcarry += __shfl(iz, 31, 32);
  }

  sF = wave_reduce_f(sF);
  sF2 = wave_reduce_f(sF2);
  if (lane == 0) { rowF[b] = sF; rowF2[b] = sF2; }
}

// ---------------------------------------------------------------------------
// Kernel 2: one wave per (b, m-tile). For each 16x16 (m,n) tile in the band,
// build term(m,n) = G_m^2*1 + 1*G_n^2 + (-2 G_m)*G_n via one
// V_WMMA_F32_16X16X4_F32 (exact fp32), then apply the pair predicate
//   w = (1 <= n-m <= lmax) && (H[m]==H[n])
// branchlessly on the f32 accumulator elements. Per-tile partials -> ws.
// ---------------------------------------------------------------------------
__global__ void k_pairs(const float* __restrict__ G, const int* __restrict__ H,
                        const int* __restrict__ lmax_ptr,
                        float* __restrict__ partL, float* __restrict__ partC) {
  const int lane = threadIdx.x & 31;
  const int wid = threadIdx.x >> 5;
  const int t = blockIdx.x * 4 + wid;          // 0 .. TOTTILES-1
  const int b = t >> 7;                        // / NTILES
  const int mtile = t & (NTILES - 1);
  const int m0 = mtile * 16;
  const long base = (long)b * SS;

  int lmax = *lmax_ptr;
  if (lmax > SS) lmax = SS;

  const int hi = lane >> 4;                    // 0: lanes 0-15, 1: lanes 16-31
  const int nl = lane & 15;

  // A matrix (16x4 f32), fixed for this wave. Direct per-lane load (no LDS):
  // ISA layout: VGPR0 = K=0 (lanes 0-15) / K=2 (lanes 16-31)
  //             VGPR1 = K=1 (lanes 0-15) / K=3 (lanes 16-31), M = lane&15
  const float gm = G[base + m0 + nl];
  v2f A;
  A.x = (hi == 0) ? gm * gm : -2.0f * gm;      // K=0 : G_m^2   | K=2 : -2 G_m
  A.y = (hi == 0) ? 1.0f : 0.0f;               // K=1 : 1       | K=3 : 0

  // H[m] values for this lane's 8 accumulator rows (m = m0 + 8*hi + r)
  const int mrow_base = m0 + 8 * hi;
  int hm[8];
  #pragma unroll
  for (int r = 0; r < 8; ++r) hm[r] = H[base + mrow_base + r];

  int jend = (m0 + 15 + lmax) / 16;
  if (jend > NTILES - 1) jend = NTILES - 1;

  float loss = 0.0f, cnt = 0.0f;

  auto body = [&](int j) {
    const int n0 = j * 16;
    __builtin_prefetch(&G[base + n0 + 32], 0, 3);   // speculative, next tiles

    // Per-lane direct loads; lanes 0-15 and 16-31 hit the same 16 words.
    const float gn = G[base + n0 + nl];
    const int hn = H[base + n0 + nl];

    // B matrix (4x16 f32): VGPR0 = K=0 row (1.0) / K=2 row (G_n)
    //                      VGPR1 = K=1 row (G_n^2) / K=3 row (0), N = lane&15
    v2f Bm;
    Bm.x = (hi == 0) ? 1.0f : gn;
    Bm.y = (hi == 0) ? gn * gn : 0.0f;

    v8f C = {};
    C = __builtin_amdgcn_wmma_f32_16x16x4_f32(
        /*neg_a=*/false, A, /*neg_b=*/false, Bm,
        /*c_mod=*/(short)0, C, /*reuse_a=*/false, /*reuse_b=*/false);

    // C/D layout: lane L<16 -> N=L, M=r; lane L>=16 -> N=L-16, M=r+8
    const int gap0 = (n0 + nl) - mrow_base;         // gap at r=0
    #pragma unroll
    for (int r = 0; r < 8; ++r) {
      const int gap = gap0 - r;
      const float w =
          ((gap >= 1) & (gap <= lmax) & (hm[r] == hn)) ? 1.0f : 0.0f;
      loss = fmaf(w, C[r], loss);
      cnt += w;
    }
  };

  int j = mtile;
  for (; j + 1 <= jend; j += 2) { body(j); body(j + 1); }
  if (j <= jend) body(j);

  loss = wave_reduce_f(loss);
  cnt = wave_reduce_f(cnt);
  if (lane == 0) { partL[t] = loss; partC[t] = cnt; }
}

// ---------------------------------------------------------------------------
// Kernel 3: deterministic fixed-order reduction of tile partials + flow stats.
// out layout (27 f32): [loss][total x8][num_subtraj x8][normalized x8][mean][std]
// ---------------------------------------------------------------------------
__global__ void k_final(const float* __restrict__ partL, const float* __restrict__ partC,
                        const float* __restrict__ rowF, const float* __restrict__ rowF2,
                        float* __restrict__ out) {
  const int lane = threadIdx.x;
  float norm = 0.0f;
  if (lane < BB) {
    float tot = 0.0f, cn = 0.0f;
    for (int j = 0; j < NTILES; ++j) {
      tot += partL[lane * NTILES + j];
      cn += partC[lane * NTILES + j];
    }
    const float num = fmaxf(cn, 1.0f);
    norm = tot / num;
    out[1 + lane] = tot;
    out[9 + lane] = num;
    out[17 + lane] = norm;
  }
  if (lane == 0) {
    float lsum = 0.0f;
    #pragma unroll
    for (int bb = 0; bb < BB; ++bb) lsum += __shfl(norm, bb, 32);
    out[0] = lsum / (float)BB;
    float sF = 0.0f, sF2 = 0.0f;
    for (int bb = 0; bb < BB; ++bb) { sF += rowF[bb]; sF2 += rowF2[bb]; }
    const float N = (float)(BB * SS);
    const float mean = sF / N;
    const float var = (sF2 - N * mean * mean) / (N - 1.0f);
    out[25] = mean;
    out[26] = sqrtf(fmaxf(var, 0.0f));
  }
}

// ---------------------------------------------------------------------------
extern "C" void kernel_launch(void* const* d_in, const int* in_sizes, int n_in,
                              void* d_out, int out_size, void* d_ws, size_t ws_size,
                              hipStream_t stream) {
  const float* lp = (const float*)d_in[0];
  const float* vals = (const float*)d_in[1];
  const float* ps = (const float*)d_in[2];
  const int* mask = (const int*)d_in[3];
  const int* lmax = (const int*)d_in[4];
  float* out = (float*)d_out;

  // workspace layout (bytes)
  char* ws = (char*)d_ws;
  float* G = (float*)(ws);                                // B*S f32   (64 KB)
  int* H = (int*)(ws + (size_t)BB * SS * 4);              // B*S i32   (64 KB)
  float* partL = (float*)(ws + (size_t)2 * BB * SS * 4);  // 1024 f32
  float* partC = partL + TOTTILES;                        // 1024 f32
  float* rowF = partC + TOTTILES;                         // 8 f32
  float* rowF2 = rowF + BB;                               // 8 f32

  k_prep<<<BB, 32, 0, stream>>>(lp, vals, ps, mask, G, H, rowF, rowF2);
  k_pairs<<<TOTTILES / 4, 128, 0, stream>>>(G, H, lmax, partL, partC);
  k_final<<<1, 32, 0, stream>>>(partL, partC, rowF, rowF2, out);
}